// GCNLayer_76647986365164
// MI455X (gfx1250) — compile-verified
//
#include <hip/hip_runtime.h>

#define D 128
#define EPS 1e-5f

typedef __attribute__((ext_vector_type(2))) float v2f;
typedef __attribute__((ext_vector_type(8))) float v8f;

// ---------------------------------------------------------------- zero ws ---
__global__ void zero_ws(float* __restrict__ p, int n4) {
    int i = blockIdx.x * blockDim.x + threadIdx.x;
    if (i < n4) ((float4*)p)[i] = make_float4(0.f, 0.f, 0.f, 0.f);
}

// ------------------------------------------------- edge scatter (mean agg) ---
// 32 lanes per edge; each lane handles 4 consecutive floats of the 128-wide
// row: one coalesced float4 load of feature[src], 4 atomic f32 adds to agg[dst].
__global__ void edge_scatter(const float* __restrict__ feat,
                             const int*   __restrict__ src,
                             const int*   __restrict__ dst,
                             float* __restrict__ agg,
                             float* __restrict__ deg,
                             int E_) {
    long gid  = (long)blockIdx.x * blockDim.x + threadIdx.x;
    int  e    = (int)(gid >> 5);
    int  lane = (int)(gid & 31);
    if (e >= E_) return;
    int s = src[e];
    int d = dst[e];
    const float4 v = ((const float4*)(feat + (size_t)s * D))[lane];
    float* ap = agg + (size_t)d * D + lane * 4;
    atomicAdd(ap + 0, v.x);
    atomicAdd(ap + 1, v.y);
    atomicAdd(ap + 2, v.z);
    atomicAdd(ap + 3, v.w);
    if (lane == 0) atomicAdd(deg + d, 1.0f);
}

// ------------------------------------------- fp32 WMMA GEMM: h = (agg/deg)W^T + b
// Per block: 256 threads = 8 waves, W staged in 64KB LDS.
// Per wave: 16-row strip x 128 cols, K looped in steps of 4 via
// v_wmma_f32_16x16x4_f32.  C = A * B with A = agg/deg (16x128), B = W^T,
// i.e. B[k][n] = W[n][k] (row-major W, so B frag is a contiguous float2).
__global__ void __launch_bounds__(256)
gcn_gemm_wmma(const float* __restrict__ agg,
              const float* __restrict__ deg,
              const float* __restrict__ Wm,
              const float* __restrict__ bias,
              float* __restrict__ h, int N_) {
    __shared__ float lW[D * D];   // 64 KB (WGP has 320 KB)
    const int tid = threadIdx.x;
    for (int i = tid; i < D * D / 4; i += 256)
        ((float4*)lW)[i] = ((const float4*)Wm)[i];
    __syncthreads();

    const int wave  = tid >> 5;
    const int lane  = tid & 31;
    const int m     = lane & 15;            // matrix row (A) / col (B,C)
    const int khalf = (lane >> 4) << 1;     // 0 or 2 : K sub-offset per ISA layout
    const int base  = blockIdx.x * 128 + wave * 16;

    // Clamp boundary rows for loads (EXEC must be all-1s for WMMA);
    // lane l and l+16 share m so clamping is consistent per matrix row.
    int row  = base + m;
    int rowc = row < N_ ? row : N_ - 1;
    float invd = 1.0f / fmaxf(deg[rowc], 1.0f);   // zero in-degree -> divide by 1
    const float* arow = agg + (size_t)rowc * D;

    v8f acc[8];
    #pragma unroll
    for (int t = 0; t < 8; ++t)
        #pragma unroll
        for (int j = 0; j < 8; ++j) acc[t][j] = 0.0f;

    for (int kk = 0; kk < 32; ++kk) {
        int k = kk * 4 + khalf;
        v2f a = *(const v2f*)(arow + k);    // A[m][k], A[m][k+1]
        a = a * invd;                        // fuse mean-divide into A fragment
        #pragma unroll
        for (int nt = 0; nt < 8; ++nt) {
            // B frag: b.x = W[n][k], b.y = W[n][k+1], n = nt*16 + m
            v2f b = *(const v2f*)(lW + (nt * 16 + m) * D + k);
            acc[nt] = __builtin_amdgcn_wmma_f32_16x16x4_f32(
                false, a, false, b, (short)0, acc[nt], false, false);
        }
    }

    // Epilogue: bias add + predicated store.
    // C layout: VGPR r, lane l -> row = base + r + 8*(l>=16), col = nt*16 + m.
    const int rowlo = base + ((lane >> 4) ? 8 : 0);
    #pragma unroll
    for (int nt = 0; nt < 8; ++nt) {
        float bv = bias[nt * 16 + m];
        #pragma unroll
        for (int r = 0; r < 8; ++r) {
            int orow = rowlo + r;
            if (orow < N_)
                h[(size_t)orow * D + nt * 16 + m] = acc[nt][r] + bv;
        }
    }
}

// --------------------------------------------------------- batchnorm stats ---
// blockDim = 128 (one thread per column); each block streams 256 rows
// (fully coalesced 512B per row), then 2 atomics per column per block.
__global__ void bn_stats(const float* __restrict__ h,
                         float* __restrict__ gsum,
                         float* __restrict__ gsq, int N_) {
    int col  = threadIdx.x;
    int r0   = blockIdx.x * 256;
    int rend = min(r0 + 256, N_);
    float s = 0.f, s2 = 0.f;
    for (int r = r0; r < rend; ++r) {
        float v = h[(size_t)r * D + col];
        s  += v;
        s2 += v * v;
    }
    atomicAdd(&gsum[col], s);
    atomicAdd(&gsq[col],  s2);
}

// ------------------------------------- normalize + relu + residual (inplace) ---
__global__ void bn_finalize(const float* __restrict__ feat,
                            const float* __restrict__ gsum,
                            const float* __restrict__ gsq,
                            const float* __restrict__ gamma,
                            const float* __restrict__ beta,
                            float* __restrict__ out, int N_) {
    int i = blockIdx.x * blockDim.x + threadIdx.x;   // one float4 per thread
    long n4 = (long)N_ * (D / 4);
    if (i >= n4) return;
    const float invN = 1.0f / (float)N_;
    int c0 = (i & 31) * 4;                           // (i*4) % 128
    float4 hv = ((float4*)out)[i];
    float4 fv = ((const float4*)feat)[i];
    float hr[4] = {hv.x, hv.y, hv.z, hv.w};
    float fr[4] = {fv.x, fv.y, fv.z, fv.w};
    float o[4];
    #pragma unroll
    for (int j = 0; j < 4; ++j) {
        int c = c0 + j;
        float mean = gsum[c] * invN;
        float var  = gsq[c] * invN - mean * mean;    // biased var (ddof=0)
        float val  = (hr[j] - mean) * rsqrtf(var + EPS) * gamma[c] + beta[c];
        val = fmaxf(val, 0.0f);                      // relu
        o[j] = fr[j] + val;                          // residual
    }
    ((float4*)out)[i] = make_float4(o[0], o[1], o[2], o[3]);
}

// ----------------------------------------------------------------- launch ---
extern "C" void kernel_launch(void* const* d_in, const int* in_sizes, int n_in,
                              void* d_out, int out_size, void* d_ws, size_t ws_size,
                              hipStream_t stream) {
    const float* feat  = (const float*)d_in[0];
    const int*   src   = (const int*)  d_in[1];
    const int*   dst   = (const int*)  d_in[2];
    const float* Wm    = (const float*)d_in[3];
    const float* bias  = (const float*)d_in[4];
    const float* gamma = (const float*)d_in[5];
    const float* beta  = (const float*)d_in[6];

    const int N_ = in_sizes[0] / D;
    const int E_ = in_sizes[1];

    // Workspace layout: agg[N*D] | deg[N] | gsum[128] | gsq[128]  (~51.6 MB)
    float* ws   = (float*)d_ws;
    float* agg  = ws;
    float* deg  = ws + (size_t)N_ * D;
    float* gsum = deg + N_;
    float* gsq  = gsum + D;
    float* h    = (float*)d_out;   // GEMM output, then normalized in place

    size_t zfloats = (size_t)N_ * D + N_ + 2 * D;
    int z4 = (int)((zfloats + 3) / 4);
    zero_ws<<<(z4 + 255) / 256, 256, 0, stream>>>(ws, z4);

    long sthreads = (long)E_ * 32;
    edge_scatter<<<(int)((sthreads + 255) / 256), 256, 0, stream>>>(
        feat, src, dst, agg, deg, E_);

    gcn_gemm_wmma<<<(N_ + 127) / 128, 256, 0, stream>>>(agg, deg, Wm, bias, h, N_);

    bn_stats<<<(N_ + 255) / 256, 128, 0, stream>>>(h, gsum, gsq, N_);

    int n4 = N_ * (D / 4);
    bn_finalize<<<(n4 + 255) / 256, 256, 0, stream>>>(
        feat, gsum, gsq, gamma, beta, h, N_);
}